// GraphEmbedding_58171037057074
// MI455X (gfx1250) — compile-verified
//
#include <hip/hip_runtime.h>

// CDNA5 / gfx1250: wave32, WMMA f32 16x16x4.
typedef float v2f __attribute__((ext_vector_type(2)));
typedef float v8f __attribute__((ext_vector_type(8)));

#define THREADS 256

// ---------- normalization precompute ----------
__global__ void k_zero(float* __restrict__ p, int n) {
    int i = blockIdx.x * blockDim.x + threadIdx.x;
    if (i < n) p[i] = 0.0f;
}

__global__ void k_deg(const int* __restrict__ col, const float* __restrict__ ew,
                      float* __restrict__ deg, int nE) {
    int e = blockIdx.x * blockDim.x + threadIdx.x;
    if (e < nE) atomicAdd(&deg[col[e]], ew[e]);
}

// dis[] currently holds deg without self-loop; add self-loop weight 1.
// norm_self = dis*dis = 1/deg ; dis = rsqrt(deg). deg >= 1 always.
__global__ void k_degfin(float* __restrict__ dis, float* __restrict__ nself, int n) {
    int i = blockIdx.x * blockDim.x + threadIdx.x;
    if (i < n) {
        float d = dis[i] + 1.0f;
        nself[i] = 1.0f / d;
        dis[i]   = rsqrtf(d);
    }
}

__global__ void k_norme(const int* __restrict__ row, const int* __restrict__ col,
                        const float* __restrict__ ew, const float* __restrict__ dis,
                        float* __restrict__ norm, int nE) {
    int e = blockIdx.x * blockDim.x + threadIdx.x;
    if (e < nE) norm[e] = dis[row[e]] * ew[e] * dis[col[e]];
}

// ---------- WMMA GEMM: H[n x 64] = X[n x K] @ W[K x 64], K in {16,64} ----------
// Block = 256 threads = 8 waves; each wave owns a 16-row stripe and all 4
// 16-wide column tiles. A-layout per ISA (16x4 f32): lane m=lane&15 holds row,
// kb=(lane>>4)*2 selects K pair. B/C/D striped across lanes within a VGPR.
__global__ void k_gemm_wmma(const float* __restrict__ X, const float* __restrict__ W,
                            float* __restrict__ H, int n, int K) {
    __shared__ float sW[64 * 64];  // 16 KB max (K=64)
    int tid = threadIdx.x;
    for (int i = tid; i < K * 64; i += blockDim.x) sW[i] = W[i];
    __syncthreads();

    int wave = tid >> 5;
    int lane = tid & 31;
    int rowTile = blockIdx.x * 8 + wave;
    if (rowTile * 16 >= n) return;  // wave-uniform: EXEC stays all-ones for WMMA

    int r0 = rowTile * 16;
    int m  = lane & 15;
    int kb = (lane >> 4) * 2;
    const float* arow = X + (size_t)(r0 + m) * K;

    v8f acc0 = {}, acc1 = {}, acc2 = {}, acc3 = {};
    for (int k0 = 0; k0 < K; k0 += 4) {
        v2f a;
        a.x = arow[k0 + kb];
        a.y = arow[k0 + kb + 1];
        const float* brow0 = &sW[(k0 + kb) * 64];
        const float* brow1 = &sW[(k0 + kb + 1) * 64];
        v2f b;
        b.x = brow0[m];       b.y = brow1[m];
        acc0 = __builtin_amdgcn_wmma_f32_16x16x4_f32(false, a, false, b, (short)0, acc0, false, false);
        b.x = brow0[16 + m];  b.y = brow1[16 + m];
        acc1 = __builtin_amdgcn_wmma_f32_16x16x4_f32(false, a, false, b, (short)0, acc1, false, false);
        b.x = brow0[32 + m];  b.y = brow1[32 + m];
        acc2 = __builtin_amdgcn_wmma_f32_16x16x4_f32(false, a, false, b, (short)0, acc2, false, false);
        b.x = brow0[48 + m];  b.y = brow1[48 + m];
        acc3 = __builtin_amdgcn_wmma_f32_16x16x4_f32(false, a, false, b, (short)0, acc3, false, false);
    }

    // D layout: VGPR v, lanes 0-15 -> row r0+v, lanes 16-31 -> row r0+8+v; col = m.
    int rbase = r0 + (lane >> 4) * 8;
#pragma unroll
    for (int v = 0; v < 8; v++) {
        float* hrow = H + (size_t)(rbase + v) * 64 + m;
        hrow[0]  = acc0[v];
        hrow[16] = acc1[v];
        hrow[32] = acc2[v];
        hrow[48] = acc3[v];
    }
}

// ---------- out = bias + norm_self * h  (self-loop term + bias init) ----------
__global__ void k_initout(const float* __restrict__ H, const float* __restrict__ nself,
                          const float* __restrict__ bias, float* __restrict__ out, int n) {
    int i = blockIdx.x * blockDim.x + threadIdx.x;
    if (i < n * 64) {
        int node = i >> 6;
        int c    = i & 63;
        out[i] = bias[c] + nself[node] * H[i];
    }
}

// ---------- edge scatter: out[col] += norm[e] * h[row]  (wave per edge) ----------
__global__ void k_scatter(const float* __restrict__ H, const int* __restrict__ row,
                          const int* __restrict__ col, const float* __restrict__ norm,
                          float* __restrict__ out, int nE) {
    int wid    = (blockIdx.x * blockDim.x + threadIdx.x) >> 5;
    int lane   = threadIdx.x & 31;
    int nwaves = (gridDim.x * blockDim.x) >> 5;
    for (int e = wid; e < nE; e += nwaves) {
        int   r = row[e];
        int   c = col[e];
        float w = norm[e];
        const float* hs = H   + (size_t)r * 64;
        float*       od = out + (size_t)c * 64;
        atomicAdd(&od[lane],      w * hs[lane]);       // no-return global atomics
        atomicAdd(&od[lane + 32], w * hs[lane + 32]);
    }
}

__global__ void k_relu(float* __restrict__ p, int n) {
    int i = blockIdx.x * blockDim.x + threadIdx.x;
    if (i < n) p[i] = fmaxf(p[i], 0.0f);
}

extern "C" void kernel_launch(void* const* d_in, const int* in_sizes, int n_in,
                              void* d_out, int out_size, void* d_ws, size_t ws_size,
                              hipStream_t stream) {
    const float* x    = (const float*)d_in[0];
    const int*   eidx = (const int*)d_in[1];   // JAX x64 disabled -> int32
    const float* ew   = (const float*)d_in[2];
    const float* W1   = (const float*)d_in[3];
    const float* b1   = (const float*)d_in[4];
    const float* W2   = (const float*)d_in[5];
    const float* b2   = (const float*)d_in[6];
    const float* W3   = (const float*)d_in[7];
    const float* b3   = (const float*)d_in[8];
    float*       out  = (float*)d_out;

    const int E  = in_sizes[2];
    const int K1 = in_sizes[3] / 64;       // IN_DIM = 16
    const int N  = in_sizes[0] / K1;       // 100000
    const int* row = eidx;
    const int* col = eidx + E;

    float* ws    = (float*)d_ws;
    float* dis   = ws;                       // N  (holds deg, then rsqrt(deg))
    float* nself = dis + N;                  // N
    float* norme = nself + N;                // E
    float* hbuf  = norme + E;                // N*64
    float* acc   = hbuf + (size_t)N * 64;    // N*64

    const int gN    = (N + THREADS - 1) / THREADS;
    const int gE    = (E + THREADS - 1) / THREADS;
    const int gNE   = (N * 64 + THREADS - 1) / THREADS;
    const int gGemm = ((N + 15) / 16 + 7) / 8;
    const int gScat = 2048;

    // edge-invariant normalization, computed once per call
    k_zero  <<<gN, THREADS, 0, stream>>>(dis, N);
    k_deg   <<<gE, THREADS, 0, stream>>>(col, ew, dis, E);
    k_degfin<<<gN, THREADS, 0, stream>>>(dis, nself, N);
    k_norme <<<gE, THREADS, 0, stream>>>(row, col, ew, dis, norme, E);

    // layer 1: h1 = relu(conv(x, W1, b1))  -> acc
    k_gemm_wmma<<<gGemm, THREADS, 0, stream>>>(x, W1, hbuf, N, K1);
    k_initout  <<<gNE,   THREADS, 0, stream>>>(hbuf, nself, b1, acc, N);
    k_scatter  <<<gScat, THREADS, 0, stream>>>(hbuf, row, col, norme, acc, E);
    k_relu     <<<gNE,   THREADS, 0, stream>>>(acc, N * 64);

    // layer 2: h2 = relu(conv(h1, W2, b2)) -> acc (gemm consumed acc into hbuf first)
    k_gemm_wmma<<<gGemm, THREADS, 0, stream>>>(acc, W2, hbuf, N, 64);
    k_initout  <<<gNE,   THREADS, 0, stream>>>(hbuf, nself, b2, acc, N);
    k_scatter  <<<gScat, THREADS, 0, stream>>>(hbuf, row, col, norme, acc, E);
    k_relu     <<<gNE,   THREADS, 0, stream>>>(acc, N * 64);

    // layer 3: out = conv(h2, W3, b3) (no relu), scatter straight into d_out
    k_gemm_wmma<<<gGemm, THREADS, 0, stream>>>(acc, W3, hbuf, N, 64);
    k_initout  <<<gNE,   THREADS, 0, stream>>>(hbuf, nself, b3, out, N);
    k_scatter  <<<gScat, THREADS, 0, stream>>>(hbuf, row, col, norme, out, E);
}